// DeeperGCN_25572235281181
// MI455X (gfx1250) — compile-verified
//
#include <hip/hip_runtime.h>

#define CDIM 128
#define HDIM 256
#define NLAYERS 6

typedef __attribute__((ext_vector_type(2))) float v2f;
typedef __attribute__((ext_vector_type(8))) float v8f;

__device__ inline v8f wmma4(v2f a, v2f b, v8f c) {
  // D = A(16x4,f32) x B(4x16,f32) + C(16x16,f32)
  return __builtin_amdgcn_wmma_f32_16x16x4_f32(false, a, false, b, (short)0, c, false, false);
}

// ============ standalone pre-norm (layer 0 only): hn = relu(LN(x)*s + b) ====
// one wave (32 lanes) per node, 4 channels per lane
__global__ __launch_bounds__(128) void k_prenorm(
    const float* __restrict__ x, const float* __restrict__ ls,
    const float* __restrict__ lb, float* __restrict__ hn, int n) {
  int wid = threadIdx.x >> 5, lane = threadIdx.x & 31;
  int node = blockIdx.x * 4 + wid;
  if (node >= n) return;
  const float4* xr = reinterpret_cast<const float4*>(x + (size_t)node * CDIM);
  float4 v = xr[lane];
  float s = v.x + v.y + v.z + v.w;
  for (int o = 16; o > 0; o >>= 1) s += __shfl_xor(s, o, 32);
  float mu = s * (1.0f / CDIM);
  float d0 = v.x - mu, d1 = v.y - mu, d2 = v.z - mu, d3 = v.w - mu;
  float q = d0 * d0 + d1 * d1 + d2 * d2 + d3 * d3;
  for (int o = 16; o > 0; o >>= 1) q += __shfl_xor(q, o, 32);
  float rinv = rsqrtf(q * (1.0f / CDIM) + 1e-5f);
  const float4* lsv = reinterpret_cast<const float4*>(ls);
  const float4* lbv = reinterpret_cast<const float4*>(lb);
  float4 sc = lsv[lane], bb = lbv[lane];
  float4 out;
  out.x = fmaxf(d0 * rinv * sc.x + bb.x, 0.f);
  out.y = fmaxf(d1 * rinv * sc.y + bb.y, 0.f);
  out.z = fmaxf(d2 * rinv * sc.z + bb.z, 0.f);
  out.w = fmaxf(d3 * rinv * sc.w + bb.w, 0.f);
  reinterpret_cast<float4*>(hn + (size_t)node * CDIM)[lane] = out;
}

// ============ single edge pass: ex = exp(msg*t); sum ex and msg*ex =========
// one wave per edge (32 lanes x 4 channels); softmax shift-invariance lets us
// skip the segment-max pass (logits bounded, no overflow risk in f32)
__global__ __launch_bounds__(256) void k_edge_expsum(
    const float* __restrict__ hn, const int* __restrict__ src,
    const int* __restrict__ dst, const float* __restrict__ tptr,
    float* __restrict__ denom, float* __restrict__ numer, int E) {
  int gw = (blockIdx.x * blockDim.x + threadIdx.x) >> 5;  // wave index == edge
  int e = __builtin_amdgcn_readfirstlane(gw);
  if (e >= E) return;
  int lane = threadIdx.x & 31;
  int s = src[e], d = dst[e];  // wave-uniform -> scalar loads
  float tt = tptr[0];
  int cq = lane << 2;
  float4 h = *reinterpret_cast<const float4*>(hn + (size_t)s * CDIM + cq);
  size_t off = (size_t)d * CDIM + cq;
  float m0 = fmaxf(h.x, 0.f) + 1e-7f;
  float m1 = fmaxf(h.y, 0.f) + 1e-7f;
  float m2 = fmaxf(h.z, 0.f) + 1e-7f;
  float m3 = fmaxf(h.w, 0.f) + 1e-7f;
  float e0 = __expf(m0 * tt);
  float e1 = __expf(m1 * tt);
  float e2 = __expf(m2 * tt);
  float e3 = __expf(m3 * tt);
  atomicAdd(denom + off + 0, e0);
  atomicAdd(denom + off + 1, e1);
  atomicAdd(denom + off + 2, e2);
  atomicAdd(denom + off + 3, e3);
  atomicAdd(numer + off + 0, m0 * e0);
  atomicAdd(numer + off + 1, m1 * e1);
  atomicAdd(numer + off + 2, m2 * e2);
  atomicAdd(numer + off + 3, m3 * e3);
}

// ============ fused node MLP + residual + next-layer pre-norm ==============
// block = 256 threads (8 waves), 32 nodes per block; each wave computes two
// 16-row tiles sharing one B fragment -> 2 wmma per weight-load pair
__global__ __launch_bounds__(256) void k_node_mlp(
    float* __restrict__ xcur, const float* __restrict__ hn,
    float* __restrict__ hnout,
    const float* __restrict__ numer, const float* __restrict__ denom,
    const float* __restrict__ W1, const float* __restrict__ b1,
    const float* __restrict__ mls, const float* __restrict__ mlb,
    const float* __restrict__ W2, const float* __restrict__ b2,
    const float* __restrict__ nls, const float* __restrict__ nlb,  // next-layer LN (may be null)
    int n) {
  __shared__ float Z[32][CDIM + 4];   // z tile; reused for x_new in epilogue
  __shared__ float Y[32][HDIM + 8];   // hidden tile
  int tid = threadIdx.x, wid = tid >> 5, lane = tid & 31;
  int row0 = blockIdx.x * 32;

  for (int i = tid; i < 32 * CDIM; i += 256) {
    int r = i >> 7, c = i & (CDIM - 1);
    float zv = 0.f;
    if (row0 + r < n) {
      size_t g = (size_t)(row0 + r) * CDIM + c;
      zv = numer[g] / (denom[g] + 1e-16f) + hn[g];
    }
    Z[r][c] = zv;
  }
  __syncthreads();

  int mrow = lane & 15;            // A-frag row / B-frag col
  int khalf = (lane >> 4) * 2;     // K sub-offset per lane half
  int rbase = (lane >> 4) * 8;     // D-frag row base

  // GEMM1: Y[32x256] = Z[32x128] @ W1[128x256]
  for (int t2 = 0; t2 < 2; t2++) {
    int ct = wid + t2 * 8;
    v8f acc0 = {}, acc1 = {};
    for (int kk = 0; kk < CDIM; kk += 4) {
      const float* wp = W1 + (size_t)(kk + khalf) * HDIM + ct * 16 + mrow;
      __builtin_prefetch(wp + 4 * HDIM, 0, 3);
      v2f b;
      b.x = wp[0];
      b.y = wp[HDIM];
      v2f a0, a1;
      a0.x = Z[mrow][kk + khalf];
      a0.y = Z[mrow][kk + khalf + 1];
      a1.x = Z[mrow + 16][kk + khalf];
      a1.y = Z[mrow + 16][kk + khalf + 1];
      acc0 = wmma4(a0, b, acc0);
      acc1 = wmma4(a1, b, acc1);
    }
    int col = ct * 16 + mrow;
    float bias = b1[col];
#pragma unroll
    for (int j = 0; j < 8; j++) {
      Y[rbase + j][col] = acc0[j] + bias;
      Y[16 + rbase + j][col] = acc1[j] + bias;
    }
  }
  __syncthreads();

  // per-row LayerNorm(256) + ReLU in LDS: 4 rows per wave
  for (int rr = 0; rr < 4; rr++) {
    int r = wid * 4 + rr;
    float vals[8];
    float s = 0.f;
#pragma unroll
    for (int j = 0; j < 8; j++) { vals[j] = Y[r][lane + j * 32]; s += vals[j]; }
    for (int o = 16; o > 0; o >>= 1) s += __shfl_xor(s, o, 32);
    float mu = s * (1.0f / HDIM);
    float q = 0.f;
#pragma unroll
    for (int j = 0; j < 8; j++) { float d = vals[j] - mu; q += d * d; }
    for (int o = 16; o > 0; o >>= 1) q += __shfl_xor(q, o, 32);
    float rinv = rsqrtf(q * (1.0f / HDIM) + 1e-5f);
#pragma unroll
    for (int j = 0; j < 8; j++) {
      int c = lane + j * 32;
      Y[r][c] = fmaxf((vals[j] - mu) * rinv * mls[c] + mlb[c], 0.f);
    }
  }
  __syncthreads();

  // GEMM2: U[32x128] = Y[32x256] @ W2[256x128]; residual into xcur and Z
  {
    int ct = wid;
    v8f acc0 = {}, acc1 = {};
    for (int kk = 0; kk < HDIM; kk += 4) {
      const float* wp = W2 + (size_t)(kk + khalf) * CDIM + ct * 16 + mrow;
      __builtin_prefetch(wp + 4 * CDIM, 0, 3);
      v2f b;
      b.x = wp[0];
      b.y = wp[CDIM];
      v2f a0, a1;
      a0.x = Y[mrow][kk + khalf];
      a0.y = Y[mrow][kk + khalf + 1];
      a1.x = Y[mrow + 16][kk + khalf];
      a1.y = Y[mrow + 16][kk + khalf + 1];
      acc0 = wmma4(a0, b, acc0);
      acc1 = wmma4(a1, b, acc1);
    }
    int col = ct * 16 + mrow;
    float bias = b2[col];
#pragma unroll
    for (int j = 0; j < 8; j++) {
      int r0 = rbase + j, r1 = 16 + rbase + j;
      float xn0 = 0.f, xn1 = 0.f;
      if (row0 + r0 < n) {
        size_t g = (size_t)(row0 + r0) * CDIM + col;
        xn0 = xcur[g] + acc0[j] + bias;
        xcur[g] = xn0;
      }
      if (row0 + r1 < n) {
        size_t g = (size_t)(row0 + r1) * CDIM + col;
        xn1 = xcur[g] + acc1[j] + bias;
        xcur[g] = xn1;
      }
      Z[r0][col] = xn0;
      Z[r1][col] = xn1;
    }
  }

  // fused pre-norm for the NEXT layer: hnout = relu(LN(x_new)*nls + nlb)
  if (nls != nullptr) {
    __syncthreads();
    for (int rr = 0; rr < 4; rr++) {
      int r = wid * 4 + rr;
      int c0 = lane * 4;
      float v0 = Z[r][c0], v1 = Z[r][c0 + 1], v2 = Z[r][c0 + 2], v3 = Z[r][c0 + 3];
      float s = v0 + v1 + v2 + v3;
      for (int o = 16; o > 0; o >>= 1) s += __shfl_xor(s, o, 32);
      float mu = s * (1.0f / CDIM);
      float d0 = v0 - mu, d1 = v1 - mu, d2 = v2 - mu, d3 = v3 - mu;
      float q = d0 * d0 + d1 * d1 + d2 * d2 + d3 * d3;
      for (int o = 16; o > 0; o >>= 1) q += __shfl_xor(q, o, 32);
      float rinv = rsqrtf(q * (1.0f / CDIM) + 1e-5f);
      if (row0 + r < n) {
        float4 out;
        out.x = fmaxf(d0 * rinv * nls[c0] + nlb[c0], 0.f);
        out.y = fmaxf(d1 * rinv * nls[c0 + 1] + nlb[c0 + 1], 0.f);
        out.z = fmaxf(d2 * rinv * nls[c0 + 2] + nlb[c0 + 2], 0.f);
        out.w = fmaxf(d3 * rinv * nls[c0 + 3] + nlb[c0 + 3], 0.f);
        *reinterpret_cast<float4*>(hnout + (size_t)(row0 + r) * CDIM + c0) = out;
      }
    }
  }
}

// ============ head hidden: ht = relu(xcur @ lin_w + lin_b) ============
__global__ __launch_bounds__(256) void k_head_hidden(
    const float* __restrict__ xcur, const float* __restrict__ W,
    const float* __restrict__ bias, float* __restrict__ ht, int n) {
  __shared__ float Z[32][CDIM + 4];
  int tid = threadIdx.x, wid = tid >> 5, lane = tid & 31;
  int row0 = blockIdx.x * 32;
  for (int i = tid; i < 32 * CDIM; i += 256) {
    int r = i >> 7, c = i & (CDIM - 1);
    Z[r][c] = (row0 + r < n) ? xcur[(size_t)(row0 + r) * CDIM + c] : 0.f;
  }
  __syncthreads();
  int mrow = lane & 15, khalf = (lane >> 4) * 2, rbase = (lane >> 4) * 8;
  int ct = wid;  // 8 waves x 16 cols = 128
  v8f acc0 = {}, acc1 = {};
  for (int kk = 0; kk < CDIM; kk += 4) {
    const float* wp = W + (size_t)(kk + khalf) * CDIM + ct * 16 + mrow;
    __builtin_prefetch(wp + 4 * CDIM, 0, 3);
    v2f b;
    b.x = wp[0];
    b.y = wp[CDIM];
    v2f a0, a1;
    a0.x = Z[mrow][kk + khalf];
    a0.y = Z[mrow][kk + khalf + 1];
    a1.x = Z[mrow + 16][kk + khalf];
    a1.y = Z[mrow + 16][kk + khalf + 1];
    acc0 = wmma4(a0, b, acc0);
    acc1 = wmma4(a1, b, acc1);
  }
  int col = ct * 16 + mrow;
  float bv = bias[col];
#pragma unroll
  for (int j = 0; j < 8; j++) {
    int r0 = rbase + j, r1 = 16 + rbase + j;
    if (row0 + r0 < n)
      ht[(size_t)(row0 + r0) * CDIM + col] = fmaxf(acc0[j] + bv, 0.f);
    if (row0 + r1 < n)
      ht[(size_t)(row0 + r1) * CDIM + col] = fmaxf(acc1[j] + bv, 0.f);
  }
}

// ============ head out: out = ht @ lin_w_out + lin_b_out (OUT=2) ============
__global__ __launch_bounds__(256) void k_head_out(
    const float* __restrict__ ht, const float* __restrict__ W,
    const float* __restrict__ bias, float* __restrict__ out, int n) {
  int idx = blockIdx.x * blockDim.x + threadIdx.x;
  if (idx >= n * 2) return;
  int node = idx >> 1, o = idx & 1;
  const float* hr = ht + (size_t)node * CDIM;
  float s = bias[o];
#pragma unroll 4
  for (int c = 0; c < CDIM; c++) s += hr[c] * W[c * 2 + o];
  out[idx] = s;
}

extern "C" void kernel_launch(void* const* d_in, const int* in_sizes, int n_in,
                              void* d_out, int out_size, void* d_ws, size_t ws_size,
                              hipStream_t stream) {
  const float* x        = (const float*)d_in[0];
  const int* edge_index = (const int*)d_in[1];
  const float* ln_scale = (const float*)d_in[2];
  const float* ln_bias  = (const float*)d_in[3];
  const float* tparam   = (const float*)d_in[4];
  const float* w1       = (const float*)d_in[5];
  const float* b1       = (const float*)d_in[6];
  const float* mln_s    = (const float*)d_in[7];
  const float* mln_b    = (const float*)d_in[8];
  const float* w2       = (const float*)d_in[9];
  const float* b2       = (const float*)d_in[10];
  const float* lin_w    = (const float*)d_in[11];
  const float* lin_b    = (const float*)d_in[12];
  const float* lin_wo   = (const float*)d_in[13];
  const float* lin_bo   = (const float*)d_in[14];

  int N = in_sizes[0] / CDIM;
  int E = in_sizes[1] / 2;
  const int* src = edge_index;
  const int* dst = edge_index + E;

  size_t NC = (size_t)N * CDIM;
  float* xcur  = (float*)d_ws;
  float* hn    = xcur + NC;
  float* denom = hn + NC;
  float* numer = denom + NC;

  hipMemcpyAsync(xcur, x, NC * sizeof(float), hipMemcpyDeviceToDevice, stream);

  int edge_blocks = (E + 7) / 8;          // one wave per edge, 8 waves/block
  int tile_blocks = (N + 31) / 32;

  // layer-0 pre-norm (subsequent layers fused into k_node_mlp epilogue)
  k_prenorm<<<(N + 3) / 4, 128, 0, stream>>>(xcur, ln_scale, ln_bias, hn, N);

  for (int l = 0; l < NLAYERS; l++) {
    hipMemsetAsync(denom, 0, 2 * NC * sizeof(float), stream);
    k_edge_expsum<<<edge_blocks, 256, 0, stream>>>(
        hn, src, dst, tparam + l, denom, numer, E);
    const float* nls = (l + 1 < NLAYERS) ? ln_scale + (l + 1) * CDIM : nullptr;
    const float* nlb = (l + 1 < NLAYERS) ? ln_bias + (l + 1) * CDIM : nullptr;
    k_node_mlp<<<tile_blocks, 256, 0, stream>>>(
        xcur, hn, hn, numer, denom,
        w1 + (size_t)l * CDIM * HDIM, b1 + (size_t)l * HDIM,
        mln_s + (size_t)l * HDIM, mln_b + (size_t)l * HDIM,
        w2 + (size_t)l * HDIM * CDIM, b2 + (size_t)l * CDIM,
        nls, nlb, N);
  }

  // head (reuse denom as hidden buffer)
  k_head_hidden<<<tile_blocks, 256, 0, stream>>>(xcur, lin_w, lin_b, denom, N);
  k_head_out<<<(N * 2 + 255) / 256, 256, 0, stream>>>(denom, lin_wo, lin_bo,
                                                      (float*)d_out, N);
}